// RotaryPositionalEmbedding_11458972746333
// MI455X (gfx1250) — compile-verified
//
#include <hip/hip_runtime.h>

// RoPE, (8,32,4096,128) f32, memory-bound streaming kernel for gfx1250.
// One wave32 per token: lane L handles dim elements [4L..4L+3] = pairs k=2L, 2L+1.

typedef __attribute__((ext_vector_type(4))) float v4f;

#define LOG2_THETA 13.287712379549449f   // log2(10000)
#define INV_2PI    0.15915494309189535f  // 1/(2*pi)

__global__ __launch_bounds__(256) void rope_fwd_kernel(
    const float* __restrict__ x,
    const int*   __restrict__ token_pos,
    float*       __restrict__ out,
    int nTokens)
{
    const int tid   = blockIdx.x * 256 + threadIdx.x;  // one float4 per thread
    const int lane  = threadIdx.x & 31;
    const int token = tid >> 5;                         // wave-uniform (wave32)
    if (token >= nTokens) return;

    // Wave-uniform position -> force scalar (SMEM) load path.
    const int   p  = token_pos[__builtin_amdgcn_readfirstlane(token)];
    const float pf = (float)p;

    // inv_freq in *revolutions*: theta^(-2k/128) / (2*pi), for k0=2*lane, k1=2*lane+1.
    // exponent(k) = -k * log2(theta)/64  ->  k0: -lane*log2(theta)/32
    const float f0 = __builtin_amdgcn_exp2f((float)lane          * (-LOG2_THETA * (1.0f / 32.0f))) * INV_2PI;
    const float f1 = __builtin_amdgcn_exp2f((float)(2 * lane + 1) * (-LOG2_THETA * (1.0f / 64.0f))) * INV_2PI;

    // Angle in revolutions; exact periodic reduction via round-to-nearest-even.
    const float a0 = pf * f0;
    const float a1 = pf * f1;
    const float r0 = a0 - __builtin_rintf(a0);   // v_rndne_f32 + v_sub
    const float r1 = a1 - __builtin_rintf(a1);

    // CDNA5 hardware trig: v_sin/v_cos take revolutions (sin(2*pi*x)).
    const float c0 = __builtin_amdgcn_cosf(r0);
    const float s0 = __builtin_amdgcn_sinf(r0);
    const float c1 = __builtin_amdgcn_cosf(r1);
    const float s1 = __builtin_amdgcn_sinf(r1);

    // 128-bit non-temporal load: (even0, odd0, even1, odd1)
    const v4f v = __builtin_nontemporal_load(&((const v4f*)x)[tid]);

    v4f o;
    o.x = c0 * v.x - s0 * v.y;   // out_even(k0)
    o.y = s0 * v.x + c0 * v.y;   // out_odd (k0)
    o.z = c1 * v.z - s1 * v.w;   // out_even(k1)
    o.w = s1 * v.z + c1 * v.w;   // out_odd (k1)

    // 128-bit non-temporal store: single-pass stream, keep L2 clean.
    __builtin_nontemporal_store(o, &((v4f*)out)[tid]);
}

extern "C" void kernel_launch(void* const* d_in, const int* in_sizes, int n_in,
                              void* d_out, int out_size, void* d_ws, size_t ws_size,
                              hipStream_t stream)
{
    const float* x   = (const float*)d_in[0];
    const int*   tp  = (const int*)d_in[1];   // token_positions (harness delivers int*)
    float*       out = (float*)d_out;

    const long long n     = (long long)in_sizes[0];   // 8*32*4096*128 floats
    const int nTokens     = (int)(n / 128);
    const int nVec4       = (int)(n / 4);
    const int blocks      = (nVec4 + 255) / 256;      // exact for power-of-two sizes

    rope_fwd_kernel<<<blocks, 256, 0, stream>>>(x, tp, out, nTokens);
}